// SE3Parameter_58256936403585
// MI455X (gfx1250) — compile-verified
//
#include <hip/hip_runtime.h>
#include <math.h>

typedef float v2f __attribute__((ext_vector_type(2)));
typedef float v4f __attribute__((ext_vector_type(4)));
typedef float v8f __attribute__((ext_vector_type(8)));

#define EPS_THETA 1e-6f

// Build the 4x4 SE3 matrix T = [[R, t],[0,0,0,1]] for pose p (Rodrigues).
__device__ __forceinline__ void se3_T(const float* __restrict__ xi, long p, float T[16]) {
    const float* x = xi + p * 6;
    const float t0 = x[0], t1 = x[1], t2 = x[2];
    const float w0 = x[3], w1 = x[4], w2 = x[5];
    const float th2 = w0 * w0 + w1 * w1 + w2 * w2;
    const float th  = sqrtf(th2);
    const bool  sm  = th < EPS_THETA;
    const float inv = sm ? 1.0f : (1.0f / th);
    const float k0 = w0 * inv, k1 = w1 * inv, k2 = w2 * inv;
    const float s = sinf(th);
    const float c = 1.0f - cosf(th);
    const float kk = k0 * k0 + k1 * k1 + k2 * k2;   // K^2 = k k^T - (k.k) I
    float r00 = 1.0f + c * (k0 * k0 - kk);
    float r01 = -s * k2 + c * (k0 * k1);
    float r02 =  s * k1 + c * (k0 * k2);
    float r10 =  s * k2 + c * (k1 * k0);
    float r11 = 1.0f + c * (k1 * k1 - kk);
    float r12 = -s * k0 + c * (k1 * k2);
    float r20 = -s * k1 + c * (k2 * k0);
    float r21 =  s * k0 + c * (k2 * k1);
    float r22 = 1.0f + c * (k2 * k2 - kk);
    if (sm) {   // R_small = I + skew(omega)
        r00 = 1.0f; r01 = -w2;  r02 = w1;
        r10 = w2;   r11 = 1.0f; r12 = -w0;
        r20 = -w1;  r21 = w0;   r22 = 1.0f;
    }
    T[0] = r00; T[1] = r01; T[2]  = r02; T[3]  = t0;
    T[4] = r10; T[5] = r11; T[6]  = r12; T[7]  = t1;
    T[8] = r20; T[9] = r21; T[10] = r22; T[11] = t2;
    T[12] = 0.0f; T[13] = 0.0f; T[14] = 0.0f; T[15] = 1.0f;
}

// Main kernel: 1 pose per thread for Rodrigues; 1 wave = 32 poses; the 4x4
// composes are done 4-at-a-time on v_wmma_f32_16x16x4_f32 (A = 4 stacked T's
// (16x4), B = 4 side-by-side P's (4x16); diagonal 4x4 blocks of D are the
// products). Requires gridDim.x*128 poses exactly (tail kernel handles rest).
__global__ __launch_bounds__(128) void se3_wmma_kernel(
    const float* __restrict__ xi, const float* __restrict__ poses,
    float* __restrict__ out) {
    __shared__ v4f ldsT[4][128];   // per-wave: 32 poses x 16 floats
    __shared__ v4f ldsP[4][128];
    __shared__ v4f ldsO[4][128];

    const int tid  = threadIdx.x;
    const int wave = tid >> 5;
    const int lane = tid & 31;
    const long pose = (long)blockIdx.x * 128 + tid;       // this lane's pose
    const long p0   = (long)blockIdx.x * 128 + wave * 32; // wave's first pose

    // ---- Stage P: coalesced non-temporal b128 loads (streaming, bypass LRU) ----
    const v4f* Pg = (const v4f*)(poses + p0 * 16);
    #pragma unroll
    for (int r = 0; r < 4; ++r) {
        const int c = r * 32 + lane;
        ldsP[wave][c] = __builtin_nontemporal_load(Pg + c);
    }

    // ---- Rodrigues for this lane's pose, stage T into LDS ----
    float T[16];
    se3_T(xi, pose, T);
    #pragma unroll
    for (int r = 0; r < 4; ++r) {
        v4f row;
        row.x = T[r * 4 + 0]; row.y = T[r * 4 + 1];
        row.z = T[r * 4 + 2]; row.w = T[r * 4 + 3];
        ldsT[wave][lane * 4 + r] = row;
    }
    __syncthreads();

    const float* Tw = (const float*)ldsT[wave];
    const float* Pw = (const float*)ldsP[wave];
    float*       Ow = (float*)ldsO[wave];

    // Fragment coordinates for wave32 f32 WMMA:
    //  A (16x4): lanes 0-15 hold rows M=lane, VGPR {0,1} = K {0,1};
    //            lanes 16-31 same rows, VGPR {0,1} = K {2,3}.
    //  B (4x16): lanes 0-15 hold cols N=lane, VGPR {0,1} = K {0,1};
    //            lanes 16-31 same cols, K {2,3}.
    //  D (16x16): lanes 0-15 VGPR v = D[v][lane]; lanes 16-31 VGPR v = D[v+8][lane-16].
    const int  half   = lane >> 4;          // which K-pair / which row half of D
    const int  a      = (lane & 15) >> 2;   // sub-block (pose) index 0..3
    const int  ri     = lane & 3;           // row (for A) / column (for B, D)
    const bool active = (half == (a >> 1)); // does this lane own a diagonal block?
    const bool hi     = (a & 1) != 0;       // block rows live in VGPRs 4..7 (vs 0..3)

    #pragma unroll
    for (int w = 0; w < 8; ++w) {
        const int l0 = w * 4;           // first of the 4 poses in this WMMA
        v2f A, B;
        // A[lane&15][k] = T_{l0+a}[ri][k], k = 2*half + {0,1} (contiguous)
        const float* tp = Tw + (l0 + a) * 16 + ri * 4 + 2 * half;
        A.x = tp[0];
        A.y = tp[1];
        // B[k][lane&15] = P_{l0+a}[k][ri], k = 2*half + {0,1} (stride 4)
        const float* pp = Pw + (l0 + a) * 16 + half * 8 + ri;
        B.x = pp[0];
        B.y = pp[4];

        v8f C = {0.f, 0.f, 0.f, 0.f, 0.f, 0.f, 0.f, 0.f};
        C = __builtin_amdgcn_wmma_f32_16x16x4_f32(
                /*neg_a=*/false, A, /*neg_b=*/false, B,
                /*c_mod=*/(short)0, C, /*reuse_a=*/false, /*reuse_b=*/false);

        // Scatter the diagonal 4x4 block owned by this lane (if any) to LDS.
        // Constant-index extractions + one select (hi) instead of dynamic
        // vector indexing (which lowered to giant v_cndmask cascades).
        if (active) {
            const float e0 = hi ? C[4] : C[0];
            const float e1 = hi ? C[5] : C[1];
            const float e2 = hi ? C[6] : C[2];
            const float e3 = hi ? C[7] : C[3];
            float* o = Ow + (l0 + a) * 16 + ri;
            o[0]  = e0;
            o[4]  = e1;
            o[8]  = e2;
            o[12] = e3;
        }
    }
    __syncthreads();

    // ---- Coalesced non-temporal b128 stores of the 32 results ----
    v4f* Og = (v4f*)(out + p0 * 16);
    #pragma unroll
    for (int r = 0; r < 4; ++r) {
        const int c = r * 32 + lane;
        __builtin_nontemporal_store(ldsO[wave][c], Og + c);
    }
}

// Scalar fallback for a non-multiple-of-128 tail (not hit for N=2,000,000).
__global__ void se3_tail_kernel(const float* __restrict__ xi,
                                const float* __restrict__ poses,
                                float* __restrict__ out, int start, int n) {
    const int p = start + blockIdx.x * blockDim.x + threadIdx.x;
    if (p >= n) return;
    float T[16];
    se3_T(xi, p, T);
    const float* P = poses + (long)p * 16;
    float* O = out + (long)p * 16;
    #pragma unroll
    for (int i = 0; i < 4; ++i)
        #pragma unroll
        for (int k = 0; k < 4; ++k) {
            float acc = 0.0f;
            #pragma unroll
            for (int j = 0; j < 4; ++j) acc = fmaf(T[i * 4 + j], P[j * 4 + k], acc);
            O[i * 4 + k] = acc;
        }
}

extern "C" void kernel_launch(void* const* d_in, const int* in_sizes, int n_in,
                              void* d_out, int out_size, void* d_ws, size_t ws_size,
                              hipStream_t stream) {
    const float* xi    = (const float*)d_in[0];   // (N,6)  float32
    const float* poses = (const float*)d_in[1];   // (N,4,4) float32
    float* out = (float*)d_out;                   // (N,4,4) float32

    const int n = in_sizes[0] / 6;                // N = 2,000,000
    const int nmain = (n / 128) * 128;            // 128 poses per block (4 waves x 32)
    if (nmain > 0) {
        se3_wmma_kernel<<<nmain / 128, 128, 0, stream>>>(xi, poses, out);
    }
    const int rem = n - nmain;
    if (rem > 0) {
        se3_tail_kernel<<<(rem + 127) / 128, 128, 0, stream>>>(xi, poses, out, nmain, n);
    }
}